// Aggregator_9079560864591
// MI455X (gfx1250) — compile-verified
//
#include <hip/hip_runtime.h>
#include <hip/hip_bf16.h>

typedef __attribute__((ext_vector_type(2))) float v2f;
typedef __attribute__((ext_vector_type(8))) float v8f;

#define IN_DIM 128
#define OUT_DIM 128

// ---------------------------------------------------------------------------
// Kernel 1: zero the N_h accumulator (d_ws is poisoned by the harness)
// ---------------------------------------------------------------------------
__global__ __launch_bounds__(256) void zero_kernel(float4* __restrict__ p, int n4) {
    int i = blockIdx.x * 256 + threadIdx.x;
    if (i < n4) p[i] = make_float4(0.f, 0.f, 0.f, 0.f);
}

// ---------------------------------------------------------------------------
// Kernel 2: edge scatter.  One wave32 per edge: lane L handles float4 chunk L
// of the 128-float row.  unsafeAtomicAdd -> native global_atomic_add_f32
// (no-return form, STOREcnt) into the L2-resident N_h buffer.
// ---------------------------------------------------------------------------
__global__ __launch_bounds__(256) void scatter_kernel(
    const float* __restrict__ E,
    const float* __restrict__ att,
    const int*   __restrict__ src,
    const int*   __restrict__ dst,
    float* __restrict__ NH,
    int n_edges)
{
    int e = blockIdx.x * 8 + (threadIdx.x >> 5);   // 8 edges (waves) per block
    if (e >= n_edges) return;
    int lane = threadIdx.x & 31;

    int s = src[e];
    int d = dst[e];
    float a = att[e];

    const float4* erow = (const float4*)(E + (size_t)s * IN_DIM);
    float4 v = erow[lane];

    float* orow = NH + (size_t)d * IN_DIM + lane * 4;
    unsafeAtomicAdd(orow + 0, v.x * a);
    unsafeAtomicAdd(orow + 1, v.y * a);
    unsafeAtomicAdd(orow + 2, v.z * a);
    unsafeAtomicAdd(orow + 3, v.w * a);
}

// ---------------------------------------------------------------------------
// Kernel 3: per-16-node tile, compute
//   out = lrelu((e+Nh)@W1 + b1) + lrelu((e*Nh)@W2 + b2)
// with V_WMMA_F32_16X16X4_F32, K=128 chained as 32 steps of K=4.
// Block = 256 threads = 8 waves; wave w owns output columns [16w, 16w+16).
// ---------------------------------------------------------------------------
#define XPITCH 132   // LDS row pitch in floats: m*132+k -> bank (4m+k)%64, conflict-free

__global__ __launch_bounds__(256) void wmma_kernel(
    const float* __restrict__ E,
    const float* __restrict__ NH,
    const float* __restrict__ W1,
    const float* __restrict__ b1,
    const float* __restrict__ W2,
    const float* __restrict__ b2,
    float* __restrict__ out)
{
    __shared__ float x1s[16 * XPITCH];
    __shared__ float x2s[16 * XPITCH];

    const int tile     = blockIdx.x;         // 16-node tile (100000/16 = 6250 exact)
    const int tid      = threadIdx.x;
    const int nodeBase = tile * 16;

    // Stage x1 = e + Nh, x2 = e * Nh for the 16x128 tile into LDS.
    for (int i = tid; i < 16 * IN_DIM; i += 256) {
        int r = i >> 7;          // row 0..15
        int c = i & 127;         // col 0..127
        size_t g = (size_t)(nodeBase + r) * IN_DIM + c;
        float e = E[g];
        float n = NH[g];
        x1s[r * XPITCH + c] = e + n;
        x2s[r * XPITCH + c] = e * n;
    }
    __syncthreads();

    const int wave = tid >> 5;               // 0..7 -> output column tile
    const int lane = tid & 31;
    const int m    = lane & 15;              // A-fragment row
    const int kp   = (lane >> 4) * 2;        // A/B fragment K sub-offset (0 or 2)
    const int col  = wave * 16 + (lane & 15);// global output column (also B N-index)

    v8f acc1 = {};
    v8f acc2 = {};

    #pragma unroll
    for (int kk = 0; kk < 32; ++kk) {
        const int kb = kk * 4;
        // A fragments (16x4) from LDS
        v2f a1, a2, bw1, bw2;
        a1.x = x1s[m * XPITCH + kb + kp];
        a1.y = x1s[m * XPITCH + kb + kp + 1];
        a2.x = x2s[m * XPITCH + kb + kp];
        a2.y = x2s[m * XPITCH + kb + kp + 1];
        // B fragments (4x16): W row-major [k][n], n = col
        bw1.x = W1[(size_t)(kb + kp) * OUT_DIM + col];
        bw1.y = W1[(size_t)(kb + kp + 1) * OUT_DIM + col];
        bw2.x = W2[(size_t)(kb + kp) * OUT_DIM + col];
        bw2.y = W2[(size_t)(kb + kp + 1) * OUT_DIM + col];

        // (neg_a, A, neg_b, B, c_mod, C, reuse_a, reuse_b)
        acc1 = __builtin_amdgcn_wmma_f32_16x16x4_f32(false, a1, false, bw1,
                                                     (short)0, acc1, false, false);
        acc2 = __builtin_amdgcn_wmma_f32_16x16x4_f32(false, a2, false, bw2,
                                                     (short)0, acc2, false, false);
    }

    const float bb1 = b1[col];
    const float bb2 = b2[col];
    const int mbase = (lane < 16) ? 0 : 8;   // C/D layout: VGPR r -> row r or r+8

    #pragma unroll
    for (int r = 0; r < 8; ++r) {
        float v1 = acc1[r] + bb1;
        v1 = fmaxf(v1, 0.01f * v1);          // leaky_relu, slope 0.01
        float v2 = acc2[r] + bb2;
        v2 = fmaxf(v2, 0.01f * v2);
        out[(size_t)(nodeBase + mbase + r) * OUT_DIM + col] = v1 + v2;
    }
}

// ---------------------------------------------------------------------------
// Launch
// ---------------------------------------------------------------------------
extern "C" void kernel_launch(void* const* d_in, const int* in_sizes, int n_in,
                              void* d_out, int out_size, void* d_ws, size_t ws_size,
                              hipStream_t stream) {
    const float* E   = (const float*)d_in[0];   // entity_embed (N_NODES x 128)
    const float* att = (const float*)d_in[1];   // (N_EDGES)
    const float* W1  = (const float*)d_in[2];   // (128 x 128)
    const float* b1  = (const float*)d_in[3];   // (128)
    const float* W2  = (const float*)d_in[4];   // (128 x 128)
    const float* b2  = (const float*)d_in[5];   // (128)
    const int*   src = (const int*)d_in[6];     // (N_EDGES)
    const int*   dst = (const int*)d_in[7];     // (N_EDGES)

    const int n_nodes = in_sizes[0] / IN_DIM;   // 100000
    const int n_edges = in_sizes[1];            // 800000

    float* NH  = (float*)d_ws;                  // n_nodes*128 floats (51.2 MB)
    float* out = (float*)d_out;

    // 1) zero N_h
    {
        int n4 = (n_nodes * IN_DIM) / 4;
        zero_kernel<<<(n4 + 255) / 256, 256, 0, stream>>>((float4*)NH, n4);
    }
    // 2) edge scatter (8 edges per 256-thread block)
    {
        int blocks = (n_edges + 7) / 8;
        scatter_kernel<<<blocks, 256, 0, stream>>>(E, att, src, dst, NH, n_edges);
    }
    // 3) WMMA node transform (100000/16 = 6250 tiles)
    {
        int tiles = n_nodes / 16;
        wmma_kernel<<<tiles, 256, 0, stream>>>(E, NH, W1, b1, W2, b2, out);
    }
}